// MMGATLayer_86242943304375
// MI455X (gfx1250) — compile-verified
//
#include <hip/hip_runtime.h>
#include <math.h>

typedef __attribute__((ext_vector_type(2))) float v2f;
typedef __attribute__((ext_vector_type(8))) float v8f;

#define B_ 8
#define N_ 2048
#define DIN 128
#define DOUT 64
#define NEG_SLOPE 0.2f
#define LN_EPS 1e-5f
#define WAVES 8           // i-tiles (waves) per block in the fused kernel
#define LPITCH 18         // padded LDS pitch (even -> 8B-aligned b64, conflict-free)

__device__ __forceinline__ float lrelu(float v) { return fmaxf(v, NEG_SLOPE * v); }

// ---------------------------------------------------------------------------
// h[M,64] = x[M,128] @ W[128,64], M = B_*N_.  One wave per 16x16 output tile.
// A frag (16x4 f32): lane<16 -> {x[row][k], x[row][k+1]}, lane>=16 -> k+2,k+3
// B frag (4x16 f32): lane<16 -> {W[k][col], W[k+1][col]}, lane>=16 -> k+2,k+3
// ---------------------------------------------------------------------------
__global__ __launch_bounds__(32) void k_gemm_h(const float* __restrict__ x,
                                               const float* __restrict__ W,
                                               float* __restrict__ h) {
  const int lane = threadIdx.x;
  const int l15  = lane & 15;
  const int half = lane >> 4;
  const int row  = blockIdx.x * 16 + l15;  // A row
  const int col  = blockIdx.y * 16 + l15;  // B col
  const float* xr = x + (size_t)row * DIN;

  v8f acc = {};
#pragma unroll 4
  for (int k = 0; k < DIN; k += 4) {
    v2f a = *(const v2f*)(xr + k + 2 * half);
    v2f b;
    b.x = W[(k + 2 * half) * DOUT + col];
    b.y = W[(k + 2 * half + 1) * DOUT + col];
    acc = __builtin_amdgcn_wmma_f32_16x16x4_f32(false, a, false, b,
                                                (short)0, acc, false, false);
  }
  // C/D layout: VGPR r holds row (r + 8*half), col = l15 within tile
  float* hr = h + ((size_t)blockIdx.x * 16 + 8 * half) * DOUT + blockIdx.y * 16 + l15;
#pragma unroll
  for (int r = 0; r < 8; ++r) hr[(size_t)r * DOUT] = acc[r];
}

// ---------------------------------------------------------------------------
// s1[row] = h[row,:] . a1 ; s2[row] = h[row,:] . a2
// ---------------------------------------------------------------------------
__global__ void k_scores(const float* __restrict__ h, const float* __restrict__ a1,
                         const float* __restrict__ a2, float* __restrict__ s1,
                         float* __restrict__ s2) {
  int row = blockIdx.x * blockDim.x + threadIdx.x;
  if (row >= B_ * N_) return;
  const float* hr = h + (size_t)row * DOUT;
  float acc1 = 0.f, acc2 = 0.f;
#pragma unroll 8
  for (int d = 0; d < DOUT; ++d) {
    float v = hr[d];
    acc1 += v * a1[d];
    acc2 += v * a2[d];
  }
  s1[row] = acc1;
  s2[row] = acc2;
}

// ---------------------------------------------------------------------------
// smax[b] = max_j s2[b, j]   (leaky_relu is monotone, so the exact softmax
// row max is lrelu(s1_i + smax[b]) -- no online rescaling needed)
// ---------------------------------------------------------------------------
__global__ void k_bmax(const float* __restrict__ s2, float* __restrict__ smax) {
  __shared__ float red[256];
  const float* p = s2 + (size_t)blockIdx.x * N_;
  float m = -1e30f;
  for (int j = threadIdx.x; j < N_; j += 256) m = fmaxf(m, p[j]);
  red[threadIdx.x] = m;
  __syncthreads();
  for (int s = 128; s > 0; s >>= 1) {
    if (threadIdx.x < s) red[threadIdx.x] = fmaxf(red[threadIdx.x], red[threadIdx.x + s]);
    __syncthreads();
  }
  if (threadIdx.x == 0) smax[blockIdx.x] = red[0];
}

// ---------------------------------------------------------------------------
// Fused: softmax(leaky_relu(s1_i + s2_j)) @ h  +  LayerNorm.
// 8 waves per block (one batch), each wave owns one 16-row i-tile.  The h
// j-tile (16x64 = one contiguous 4KB block) is staged once per block into
// LDS, transposed + padded so B fragments are single conflict-free
// ds_load_b64 reads.  Double-buffered; s2 for the batch staged in LDS once.
// ---------------------------------------------------------------------------
__global__ __launch_bounds__(WAVES * 32) void k_attn_ln(
    const float* __restrict__ h, const float* __restrict__ s1,
    const float* __restrict__ s2, const float* __restrict__ smax,
    const float* __restrict__ gamma, const float* __restrict__ beta,
    float* __restrict__ out) {
  __shared__ float lt[2][DOUT * LPITCH];  // transposed h tile: [col][j], padded
  __shared__ float s2l[N_];               // s2 for this batch

  const int tid  = threadIdx.x;
  const int lane = tid & 31;
  const int wave = tid >> 5;
  const int l15  = lane & 15;
  const int half = lane >> 4;
  const int b    = blockIdx.y;
  const int i0   = (blockIdx.x * WAVES + wave) * 16;

  const float* hb  = h + (size_t)b * N_ * DOUT;
  const float* s2b = s2 + (size_t)b * N_;

  const float s1v  = s1[(size_t)b * N_ + i0 + l15];
  const float mrow = lrelu(s1v + smax[b]);

  // stage s2 for the whole batch (8 KB)
  for (int j = tid; j < N_; j += WAVES * 32) s2l[j] = s2b[j];

  // cooperative transposed stage of one 16x64 h tile (contiguous 4KB in h)
  auto stage = [&](int jt, int buf) {
    const float4* src = (const float4*)(hb + (size_t)jt * 16 * DOUT);
    float4 v = src[tid];            // 256 threads x float4 = 1024 floats
    int j  = tid >> 4;              // row within tile
    int c0 = (tid & 15) * 4;        // starting col
    lt[buf][(c0 + 0) * LPITCH + j] = v.x;
    lt[buf][(c0 + 1) * LPITCH + j] = v.y;
    lt[buf][(c0 + 2) * LPITCH + j] = v.z;
    lt[buf][(c0 + 3) * LPITCH + j] = v.w;
  };

  stage(0, 0);
  __syncthreads();

  v8f acc[4] = {{}, {}, {}, {}};
  float psum = 0.f;
  const int NT = N_ / 16;

  for (int jt = 0; jt < NT; ++jt) {
    const int buf = jt & 1;
    if (jt + 2 < NT)  // gfx1250 global_prefetch two tiles ahead
      __builtin_prefetch(hb + (size_t)(jt + 2) * 16 * DOUT + tid * 4, 0, 1);
    if (jt + 1 < NT) stage(jt + 1, buf ^ 1);

    const int j0 = jt * 16;
    const float* tb = lt[buf];

    v2f A[4];
#pragma unroll
    for (int kk = 0; kk < 4; ++kk) {
      v2f sv = *(const v2f*)(s2l + j0 + kk * 4 + 2 * half);
      float p0 = __expf(lrelu(s1v + sv.x) - mrow);
      float p1 = __expf(lrelu(s1v + sv.y) - mrow);
      A[kk].x = p0;
      A[kk].y = p1;
      psum += p0 + p1;
    }
#pragma unroll
    for (int kk = 0; kk < 4; ++kk) {
      const int jl = kk * 4 + 2 * half;  // even -> 8B aligned
#pragma unroll
      for (int t = 0; t < 4; ++t) {
        v2f bb = *(const v2f*)(tb + (t * 16 + l15) * LPITCH + jl);
        acc[t] = __builtin_amdgcn_wmma_f32_16x16x4_f32(false, A[kk], false, bb,
                                                       (short)0, acc[t], false, false);
      }
    }
    __syncthreads();
  }

  // Row sum of exp() for row l15: halves hold disjoint j subsets.
  float rowsum = psum + __shfl_xor(psum, 16, 32);

  // Scale accumulators by 1/rowsum.  acc[t][r] = row (r+8*half), col 16t+l15.
#pragma unroll
  for (int r = 0; r < 8; ++r) {
    float inv = 1.0f / __shfl(rowsum, r + 8 * half, 32);
#pragma unroll
    for (int t = 0; t < 4; ++t) acc[t][r] *= inv;
  }

  float g[4], be[4];
#pragma unroll
  for (int t = 0; t < 4; ++t) {
    g[t]  = gamma[t * 16 + l15];
    be[t] = beta[t * 16 + l15];
  }

  float* ob = out + ((size_t)b * N_ + i0 + 8 * half) * DOUT + l15;
#pragma unroll
  for (int r = 0; r < 8; ++r) {
    float sum = 0.f, sq = 0.f;
#pragma unroll
    for (int t = 0; t < 4; ++t) {
      float v = acc[t][r];
      sum += v;
      sq  += v * v;
    }
    // reduce across the 16 lanes of this half (xor<16 stays within the half)
#pragma unroll
    for (int m = 1; m < 16; m <<= 1) {
      sum += __shfl_xor(sum, m, 32);
      sq  += __shfl_xor(sq, m, 32);
    }
    float mean = sum * (1.0f / DOUT);
    float var  = sq * (1.0f / DOUT) - mean * mean;
    float rstd = rsqrtf(var + LN_EPS);
#pragma unroll
    for (int t = 0; t < 4; ++t)
      ob[(size_t)r * DOUT + t * 16] = (acc[t][r] - mean) * rstd * g[t] + be[t];
  }
}

// ---------------------------------------------------------------------------
extern "C" void kernel_launch(void* const* d_in, const int* in_sizes, int n_in,
                              void* d_out, int out_size, void* d_ws, size_t ws_size,
                              hipStream_t stream) {
  const float* x     = (const float*)d_in[0];
  // d_in[1] = adj : unused by the layer's math, never read
  const float* W     = (const float*)d_in[2];
  const float* a1    = (const float*)d_in[3];
  const float* a2    = (const float*)d_in[4];
  const float* gamma = (const float*)d_in[5];
  const float* beta  = (const float*)d_in[6];
  float* out = (float*)d_out;

  char* ws = (char*)d_ws;
  float* h    = (float*)ws;                                    // 16384*64*4 = 4 MB
  float* s1   = (float*)(ws + (size_t)B_ * N_ * DOUT * 4);     // 64 KB
  float* s2   = s1 + (size_t)B_ * N_;                          // 64 KB
  float* smax = s2 + (size_t)B_ * N_;                          // 32 B

  k_gemm_h<<<dim3((B_ * N_) / 16, DOUT / 16), 32, 0, stream>>>(x, W, h);
  k_scores<<<(B_ * N_) / 256, 256, 0, stream>>>(h, a1, a2, s1, s2);
  k_bmax<<<B_, 256, 0, stream>>>(s2, smax);
  k_attn_ln<<<dim3(N_ / 16 / WAVES, B_), WAVES * 32, 0, stream>>>(
      h, s1, s2, smax, gamma, beta, out);
}